// SE_GATNoEdgeAttrsMask_59365037966018
// MI455X (gfx1250) — compile-verified
//
#include <hip/hip_runtime.h>
#include <math.h>

// ---- problem constants (fixed by the reference) ----
#define HH   4
#define C1V  64
#define C2V  16
#define FINV 128
#define GV   16
#define OUTV 8

typedef __attribute__((ext_vector_type(2))) float v2f;
typedef __attribute__((ext_vector_type(8))) float v8f;

// ---------------------------------------------------------------------------
// Fill helpers (graph-capture safe init of atomic buffers)
// ---------------------------------------------------------------------------
__global__ void fill_f32(float* p, float v, int n) {
    int t = blockIdx.x * blockDim.x + threadIdx.x;
    if (t < n) p[t] = v;
}
__global__ void fill_u32(unsigned* p, unsigned v, int n) {
    int t = blockIdx.x * blockDim.x + threadIdx.x;
    if (t < n) p[t] = v;
}

// ---------------------------------------------------------------------------
// Edge list: ei*mask for real edges, then self-loops appended
// edge_index is stored (2,E): row0=src at [0,E), row1=dst at [E,2E)
// ---------------------------------------------------------------------------
__global__ void build_edges(const int* __restrict__ ei, const int* __restrict__ mask,
                            int* __restrict__ src, int* __restrict__ dst, int E, int N) {
    int t = blockIdx.x * blockDim.x + threadIdx.x;
    int Etot = E + N;
    if (t >= Etot) return;
    if (t < E) {
        int m = mask[t];
        src[t] = ei[t] * m;
        dst[t] = ei[E + t] * m;
    } else {
        src[t] = t - E;
        dst[t] = t - E;
    }
}

// ---------------------------------------------------------------------------
// C[M x Nc] = A[M x K] * W[Nc x K]^T   (fp32, exact) via V_WMMA_F32_16X16X4_F32
// One wave computes one 16x16 tile of C. wave32 layouts per ISA 7.12.2:
//   A 16x4 : lanes0-15 -> M=lane, K=0..1 in v[0..1]; lanes16-31 -> K=2..3
//   B 4x16 : lanes0-15 -> N=lane, K=r(+0) in v[r]; lanes16-31 -> K=r+2
//   C 16x16: v[r], lanes0-15 -> (M=r,     N=lane)
//            v[r], lanes16-31-> (M=r+8,   N=lane-16)
// ---------------------------------------------------------------------------
__global__ void gemm_wmma_f32(const float* __restrict__ A, const float* __restrict__ W,
                              float* __restrict__ C, int M, int K, int Nc) {
    int wave = (blockIdx.x * blockDim.x + threadIdx.x) >> 5;
    int lane = threadIdx.x & 31;
    int mtiles = (M + 15) >> 4;
    int ntiles = Nc >> 4;
    int mt = wave / ntiles;
    int nt = wave - mt * ntiles;
    if (mt >= mtiles) return;

    int half  = lane >> 4;        // 0: lanes 0-15, 1: lanes 16-31
    int l15   = lane & 15;
    int arow  = mt * 16 + l15;    // A row for this lane
    if (arow >= M) arow = M - 1;  // clamp (store is guarded)
    int khalf = half * 2;         // K offset 0 or 2 within the 4-wide step
    const float* Ap = A + (size_t)arow * K;
    const float* Wp = W + (size_t)(nt * 16 + l15) * K;   // B[k][n] = W[n][k]

    v8f acc = {};
    for (int k = 0; k < K; k += 4) {
        v2f a, b;
        a.x = Ap[k + khalf];
        a.y = Ap[k + khalf + 1];
        b.x = Wp[k + khalf];
        b.y = Wp[k + khalf + 1];
        acc = __builtin_amdgcn_wmma_f32_16x16x4_f32(
            /*neg_a=*/false, a, /*neg_b=*/false, b,
            /*c_mod=*/(short)0, acc, /*reuse_a=*/false, /*reuse_b=*/false);
    }

    int crow0 = mt * 16 + half * 8;
    int ccol  = nt * 16 + l15;
#pragma unroll
    for (int r = 0; r < 8; ++r) {
        int row = crow0 + r;
        if (row < M) C[(size_t)row * Nc + ccol] = acc[r];
    }
}

// ---------------------------------------------------------------------------
// s[n,h] = dot(h[n,h,:], att_src[h,:]);  d likewise
// ---------------------------------------------------------------------------
__global__ void att_scores(const float* __restrict__ h, const float* __restrict__ asrc,
                           const float* __restrict__ adst, float* __restrict__ s,
                           float* __restrict__ d, int N, int C) {
    int t = blockIdx.x * blockDim.x + threadIdx.x;
    if (t >= N * HH) return;
    int n = t / HH, hd = t - n * HH;
    const float* hp = h + (size_t)n * HH * C + hd * C;
    const float* ap = asrc + hd * C;
    const float* bp = adst + hd * C;
    float ss = 0.f, dd = 0.f;
    for (int c = 0; c < C; ++c) {
        float v = hp[c];
        ss += v * ap[c];
        dd += v * bp[c];
    }
    s[t] = ss;
    d[t] = dd;
}

// ordered-uint encoding so atomicMax on unsigned == float max (handles negatives)
__device__ __forceinline__ unsigned enc_f32(float f) {
    unsigned u = __float_as_uint(f);
    return (u & 0x80000000u) ? ~u : (u | 0x80000000u);
}
__device__ __forceinline__ float dec_f32(unsigned u) {
    return __uint_as_float((u & 0x80000000u) ? (u ^ 0x80000000u) : ~u);
}

// alpha = leaky_relu(s[src]+d[dst], 0.2); per-dst segment max via atomicMax
__global__ void alpha_max(const int* __restrict__ src, const int* __restrict__ dst,
                          const float* __restrict__ s, const float* __restrict__ d,
                          float* __restrict__ alpha, unsigned* __restrict__ mEnc, int Etot) {
    int t = blockIdx.x * blockDim.x + threadIdx.x;
    if (t >= Etot * HH) return;
    int e = t / HH, hd = t - e * HH;
    float a = s[src[e] * HH + hd] + d[dst[e] * HH + hd];
    a = (a > 0.f) ? a : 0.2f * a;
    alpha[t] = a;
    atomicMax(&mEnc[dst[e] * HH + hd], enc_f32(a));
}

// e = exp(alpha - m[dst]) (in place over alpha); denom[dst] += e
__global__ void exp_sum(const int* __restrict__ dst, float* __restrict__ alpha,
                        const unsigned* __restrict__ mEnc, float* __restrict__ denom, int Etot) {
    int t = blockIdx.x * blockDim.x + threadIdx.x;
    if (t >= Etot * HH) return;
    int e = t / HH, hd = t - e * HH;
    float m = dec_f32(mEnc[dst[e] * HH + hd]);
    float ex = __expf(alpha[t] - m);
    alpha[t] = ex;
    atomicAdd(&denom[dst[e] * HH + hd], ex);
}

// agg[dst, ch] += (e/denom[dst]) * h[src, ch]; one thread per (edge, channel)
__global__ void aggregate(const int* __restrict__ src, const int* __restrict__ dst,
                          const float* __restrict__ ealpha, const float* __restrict__ denom,
                          const float* __restrict__ h, float* __restrict__ agg,
                          int Etot, int C) {
    int HC = HH * C;
    int t = blockIdx.x * blockDim.x + threadIdx.x;
    if (t >= Etot * HC) return;
    int e = t / HC, ch = t - e * HC;
    int hd = ch / C;
    int de = dst[e];
    float a = ealpha[e * HH + hd] / (denom[de * HH + hd] + 1e-16f);
    atomicAdd(&agg[(size_t)de * HC + ch], a * h[(size_t)src[e] * HC + ch]);
}

__global__ void bias_relu(float* __restrict__ x, const float* __restrict__ b, int total, int HC) {
    int t = blockIdx.x * blockDim.x + threadIdx.x;
    if (t >= total) return;
    float v = x[t] + b[t % HC];
    x[t] = v > 0.f ? v : 0.f;
}

__global__ void pool(const float* __restrict__ h, const int* __restrict__ batch,
                     float* __restrict__ sums, float* __restrict__ cnt, int N, int HC) {
    int t = blockIdx.x * blockDim.x + threadIdx.x;
    if (t >= N * HC) return;
    int n = t / HC, ch = t - n * HC;
    int g = batch[n];
    atomicAdd(&sums[g * HC + ch], h[t]);
    if (ch == 0) atomicAdd(&cnt[g], 1.0f);
}

__global__ void fc_kernel(const float* __restrict__ sums, const float* __restrict__ cnt,
                          const float* __restrict__ Wfc, const float* __restrict__ bfc,
                          float* __restrict__ out) {
    int t = threadIdx.x;
    if (t >= GV * OUTV) return;
    int g = t / OUTV, o = t - g * OUTV;
    float c = cnt[g];
    c = c > 1.0f ? c : 1.0f;
    const int HC = HH * C2V;
    float acc = 0.f;
    for (int ch = 0; ch < HC; ++ch)
        acc += (sums[g * HC + ch] / c) * Wfc[o * HC + ch];
    out[t] = acc + bfc[o];
}

// ---------------------------------------------------------------------------
extern "C" void kernel_launch(void* const* d_in, const int* in_sizes, int n_in,
                              void* d_out, int out_size, void* d_ws, size_t ws_size,
                              hipStream_t stream) {
    const float* x    = (const float*)d_in[0];
    const int*   ei   = (const int*)d_in[1];
    const int*   batch= (const int*)d_in[2];
    const int*   mask = (const int*)d_in[3];
    const float* W1   = (const float*)d_in[4];
    const float* as1  = (const float*)d_in[5];
    const float* ad1  = (const float*)d_in[6];
    const float* b1   = (const float*)d_in[7];
    const float* W2   = (const float*)d_in[8];
    const float* as2  = (const float*)d_in[9];
    const float* ad2  = (const float*)d_in[10];
    const float* b2   = (const float*)d_in[11];
    const float* Wfc  = (const float*)d_in[12];
    const float* bfc  = (const float*)d_in[13];
    float* out = (float*)d_out;

    const int N    = in_sizes[2];        // batch has N elements
    const int E    = in_sizes[1] / 2;    // edge_index is (2,E)
    const int Etot = E + N;
    const int HC1  = HH * C1V;           // 256
    const int HC2  = HH * C2V;           // 64

    // ---- workspace layout ----
    char* w = (char*)d_ws;
    auto alloc = [&](size_t bytes) {
        char* p = w;
        w += (bytes + 255) & ~(size_t)255;
        return p;
    };
    int*      srcA  = (int*)alloc((size_t)Etot * 4);
    int*      dstA  = (int*)alloc((size_t)Etot * 4);
    float*    h1    = (float*)alloc((size_t)N * HC1 * 4);   // reused as h2 in layer 2
    float*    agg1  = (float*)alloc((size_t)N * HC1 * 4);   // layer-1 out / layer-2 in
    float*    agg2  = (float*)alloc((size_t)N * HC2 * 4);
    float*    sbuf  = (float*)alloc((size_t)N * HH * 4);
    float*    dbuf  = (float*)alloc((size_t)N * HH * 4);
    float*    alpha = (float*)alloc((size_t)Etot * HH * 4);
    unsigned* mEnc  = (unsigned*)alloc((size_t)N * HH * 4);
    float*    denom = (float*)alloc((size_t)N * HH * 4);
    float*    sums  = (float*)alloc((size_t)GV * HC2 * 4);
    float*    cnt   = (float*)alloc((size_t)GV * 4);

    const int B = 256;
    auto grid = [&](long long n) { return dim3((unsigned)((n + B - 1) / B)); };
    const unsigned ENC_NEG_INF = 0x007FFFFFu;   // enc_f32(-inf)

    // ---- build edge list ----
    build_edges<<<grid(Etot), B, 0, stream>>>(ei, mask, srcA, dstA, E, N);

    // ================= layer 1 =================
    {
        long long waves = (long long)((N + 15) / 16) * (HC1 / 16);
        gemm_wmma_f32<<<grid(waves * 32), B, 0, stream>>>(x, W1, h1, N, FINV, HC1);
    }
    att_scores<<<grid((long long)N * HH), B, 0, stream>>>(h1, as1, ad1, sbuf, dbuf, N, C1V);

    fill_u32<<<grid((long long)N * HH), B, 0, stream>>>(mEnc, ENC_NEG_INF, N * HH);
    fill_f32<<<grid((long long)N * HH), B, 0, stream>>>(denom, 0.f, N * HH);
    fill_f32<<<grid((long long)N * HC1), B, 0, stream>>>(agg1, 0.f, N * HC1);

    alpha_max<<<grid((long long)Etot * HH), B, 0, stream>>>(srcA, dstA, sbuf, dbuf, alpha, mEnc, Etot);
    exp_sum<<<grid((long long)Etot * HH), B, 0, stream>>>(dstA, alpha, mEnc, denom, Etot);
    aggregate<<<grid((long long)Etot * HC1), B, 0, stream>>>(srcA, dstA, alpha, denom, h1, agg1, Etot, C1V);
    bias_relu<<<grid((long long)N * HC1), B, 0, stream>>>(agg1, b1, N * HC1, HC1);

    // ================= layer 2 =================
    float* h2 = h1;   // reuse
    {
        long long waves = (long long)((N + 15) / 16) * (HC2 / 16);
        gemm_wmma_f32<<<grid(waves * 32), B, 0, stream>>>(agg1, W2, h2, N, HC1, HC2);
    }
    att_scores<<<grid((long long)N * HH), B, 0, stream>>>(h2, as2, ad2, sbuf, dbuf, N, C2V);

    fill_u32<<<grid((long long)N * HH), B, 0, stream>>>(mEnc, ENC_NEG_INF, N * HH);
    fill_f32<<<grid((long long)N * HH), B, 0, stream>>>(denom, 0.f, N * HH);
    fill_f32<<<grid((long long)N * HC2), B, 0, stream>>>(agg2, 0.f, N * HC2);
    fill_f32<<<grid(GV * HC2), B, 0, stream>>>(sums, 0.f, GV * HC2);
    fill_f32<<<grid(GV), B, 0, stream>>>(cnt, 0.f, GV);

    alpha_max<<<grid((long long)Etot * HH), B, 0, stream>>>(srcA, dstA, sbuf, dbuf, alpha, mEnc, Etot);
    exp_sum<<<grid((long long)Etot * HH), B, 0, stream>>>(dstA, alpha, mEnc, denom, Etot);
    aggregate<<<grid((long long)Etot * HC2), B, 0, stream>>>(srcA, dstA, alpha, denom, h2, agg2, Etot, C2V);
    bias_relu<<<grid((long long)N * HC2), B, 0, stream>>>(agg2, b2, N * HC2, HC2);

    // ================= pool + FC =================
    pool<<<grid((long long)N * HC2), B, 0, stream>>>(agg2, batch, sums, cnt, N, HC2);
    fc_kernel<<<1, 128, 0, stream>>>(sums, cnt, Wfc, bfc, out);
}